// DAG_Tabular2_78331613544640
// MI455X (gfx1250) — compile-verified
//
#include <hip/hip_runtime.h>
#include <math.h>

typedef __attribute__((ext_vector_type(2))) float v2f;
typedef __attribute__((ext_vector_type(8))) float v8f;

#define BN     4096
#define NCN    32
#define DIMN   64
#define DEPTHN 6
#define INNERN 128
#define CATFLT 2048
#define NCONTN 64

static __device__ __forceinline__ v8f zero8() {
  v8f z = {0.f, 0.f, 0.f, 0.f, 0.f, 0.f, 0.f, 0.f};
  return z;
}

// One 16x16 fp32 tile: acc += A(16xK, row-major, lda) @ B(Kx16 slice of row-major ldb matrix)
// using V_WMMA_F32_16X16X4_F32 (fp32 in, fp32 accumulate == reference numerics).
static __device__ __forceinline__ v8f wmma_acc_f32(const float* __restrict__ A, int lda,
                                                   const float* __restrict__ B, int ldb,
                                                   int K, v8f acc, int lane) {
  const int m  = lane & 15;   // A row (M)
  const int kh = lane >> 4;   // K half-select (lanes 16-31 hold K+2,K+3)
  const int n  = lane & 15;   // B col (N)
  const float* arow = A + m * lda + 2 * kh;
  const float* bp   = B + (size_t)(2 * kh) * ldb + n;
  for (int k = 0; k < K; k += 4) {
    v2f a, b;
    a.x = arow[k];
    a.y = arow[k + 1];
    b.x = bp[(size_t)k * ldb];
    b.y = bp[(size_t)(k + 1) * ldb];
    acc = __builtin_amdgcn_wmma_f32_16x16x4_f32(false, a, false, b, (short)0, acc,
                                                false, false);
  }
  return acc;
}

// ---------------------------------------------------------------- embedding
__global__ __launch_bounds__(256) void embed_kernel(const int* __restrict__ x_categ,
                                                    const float* __restrict__ embed,
                                                    float* __restrict__ xbuf) {
  int idx = blockIdx.x * 256 + threadIdx.x;  // < B*NC*DIM = 8388608
  int dimi = idx & 63;
  int t    = (idx >> 6) & 31;
  int b    = idx >> 11;
  int cat  = x_categ[b * NCN + t];
  xbuf[idx] = embed[(size_t)(cat + 2 + 100 * t) * DIMN + dimi];
}

// ---------------------------------------------------------------- transformer
// LayerNorm xs -> hs (32 rows x 64), 128 threads.
static __device__ void layernorm_ln(const float* xs, float* hs, const float* g,
                                    const float* b, float* red, float* mu, float* rs,
                                    int tid) {
  int r = tid >> 2, q = tid & 3;
  float s = 0.f, s2 = 0.f;
  const float* xr = xs + r * 64 + q * 16;
  for (int j = 0; j < 16; ++j) { float v = xr[j]; s += v; s2 += v * v; }
  red[r * 8 + q] = s;
  red[r * 8 + 4 + q] = s2;
  __syncthreads();
  if (tid < 32) {
    float ss  = red[tid * 8] + red[tid * 8 + 1] + red[tid * 8 + 2] + red[tid * 8 + 3];
    float ss2 = red[tid * 8 + 4] + red[tid * 8 + 5] + red[tid * 8 + 6] + red[tid * 8 + 7];
    float m   = ss * (1.0f / 64.0f);
    float var = ss2 * (1.0f / 64.0f) - m * m;
    mu[tid] = m;
    rs[tid] = rsqrtf(var + 1e-5f);
  }
  __syncthreads();
  for (int j = 0; j < 16; ++j) {
    int idx = tid * 16 + j;
    int row = idx >> 6, col = idx & 63;
    hs[idx] = (xs[idx] - mu[row]) * rs[row] * g[col] + b[col];
  }
  __syncthreads();
}

__global__ __launch_bounds__(128) void transformer_kernel(
    float* __restrict__ xbuf, const float* __restrict__ ln1_g,
    const float* __restrict__ ln1_b, const float* __restrict__ wqkv,
    const float* __restrict__ wo, const float* __restrict__ bo,
    const float* __restrict__ ln2_g, const float* __restrict__ ln2_b,
    const float* __restrict__ wff1, const float* __restrict__ bff1,
    const float* __restrict__ wff2, const float* __restrict__ bff2) {
  extern __shared__ float smem[];
  float* xs  = smem;           // 2048
  float* hs  = xs + 2048;      // 2048
  float* qs  = hs + 2048;      // 12288 : qkv [32][384], later act [32][256]
  float* os  = qs + 12288;     // 4096  : attn out [32][128]
  float* red = os + 4096;      // 256
  float* mu  = red + 256;      // 32
  float* rs  = mu + 32;        // 32

  const int tid  = threadIdx.x;
  const int wave = tid >> 5;
  const int lane = tid & 31;
  const int kh   = lane >> 4;
  const int ncol = lane & 15;

  float* xg = xbuf + (size_t)blockIdx.x * CATFLT;
  for (int i = tid; i < 2048; i += 128) xs[i] = xg[i];
  __syncthreads();

  for (int d = 0; d < DEPTHN; ++d) {
    const float* Wqkv = wqkv + (size_t)d * 64 * 384;
    const float* Wo   = wo + (size_t)d * 128 * 64;
    const float* Bo   = bo + d * 64;
    const float* W1   = wff1 + (size_t)d * 64 * 512;
    const float* B1   = bff1 + d * 512;
    const float* W2   = wff2 + (size_t)d * 256 * 64;
    const float* B2   = bff2 + d * 64;

    layernorm_ln(xs, hs, ln1_g + d * 64, ln1_b + d * 64, red, mu, rs, tid);

    // qkv = hs @ Wqkv : [32,64]x[64,384] -> 2x24 tiles
    for (int t = wave; t < 48; t += 4) {
      int mt = t & 1, nt = t >> 1;
      v8f acc = zero8();
      acc = wmma_acc_f32(hs + mt * 16 * 64, 64, Wqkv + nt * 16, 384, 64, acc, lane);
      for (int r = 0; r < 8; ++r)
        qs[(mt * 16 + r + 8 * kh) * 384 + nt * 16 + ncol] = acc[r];
    }
    __syncthreads();

    // attention: 8 heads x 32 rows = 256 tasks (VALU; ~12% of layer FLOPs)
    for (int task = tid; task < 256; task += 128) {
      int h = task >> 5, i = task & 31;
      const float* qp = qs + i * 384 + h * 16;
      float s[32];
      float mx = -1e30f;
      for (int j = 0; j < 32; ++j) {
        const float* kp = qs + j * 384 + 128 + h * 16;
        float a = 0.f;
        for (int dd = 0; dd < 16; ++dd) a += qp[dd] * kp[dd];
        a *= 0.25f;  // DH^-0.5
        s[j] = a;
        mx = fmaxf(mx, a);
      }
      float denom = 0.f;
      for (int j = 0; j < 32; ++j) { s[j] = expf(s[j] - mx); denom += s[j]; }
      float inv = 1.0f / denom;
      for (int dd = 0; dd < 16; ++dd) {
        float a = 0.f;
        for (int j = 0; j < 32; ++j) a += s[j] * qs[j * 384 + 256 + h * 16 + dd];
        os[i * 128 + h * 16 + dd] = a * inv;
      }
    }
    __syncthreads();

    // x += os @ Wo + Bo : 2x4 tiles, K=128
    for (int t = wave; t < 8; t += 4) {
      int mt = t & 1, nt = t >> 1;
      v8f acc = zero8();
      acc = wmma_acc_f32(os + mt * 16 * 128, 128, Wo + nt * 16, 64, 128, acc, lane);
      for (int r = 0; r < 8; ++r) {
        int row = mt * 16 + r + 8 * kh, col = nt * 16 + ncol;
        xs[row * 64 + col] += acc[r] + Bo[col];
      }
    }
    __syncthreads();

    layernorm_ln(xs, hs, ln2_g + d * 64, ln2_b + d * 64, red, mu, rs, tid);

    // GEGLU: act = (hs@W1[:, :256]+b) * gelu(hs@W1[:, 256:]+b) ; 2x16 tiles
    for (int t = wave; t < 32; t += 4) {
      int mt = t & 1, nt = t >> 1;
      v8f aa = zero8(), gg = zero8();
      aa = wmma_acc_f32(hs + mt * 16 * 64, 64, W1 + nt * 16, 512, 64, aa, lane);
      gg = wmma_acc_f32(hs + mt * 16 * 64, 64, W1 + 256 + nt * 16, 512, 64, gg, lane);
      for (int r = 0; r < 8; ++r) {
        int row = mt * 16 + r + 8 * kh, col = nt * 16 + ncol;
        float av  = aa[r] + B1[col];
        float gv  = gg[r] + B1[256 + col];
        float gel = 0.5f * gv * (1.0f + erff(gv * 0.7071067811865475f));
        qs[row * 256 + col] = av * gel;
      }
    }
    __syncthreads();

    // x += act @ W2 + B2 : 2x4 tiles, K=256
    for (int t = wave; t < 8; t += 4) {
      int mt = t & 1, nt = t >> 1;
      v8f acc = zero8();
      acc = wmma_acc_f32(qs + mt * 16 * 256, 256, W2 + nt * 16, 64, 256, acc, lane);
      for (int r = 0; r < 8; ++r) {
        int row = mt * 16 + r + 8 * kh, col = nt * 16 + ncol;
        xs[row * 64 + col] += acc[r] + B2[col];
      }
    }
    __syncthreads();
  }

  for (int i = tid; i < 2048; i += 128) xg[i] = xs[i];
}

// ---------------------------------------------------------------- batchnorm
__global__ __launch_bounds__(256) void bnstats_kernel(const float* __restrict__ xc,
                                                      float* __restrict__ mu,
                                                      float* __restrict__ rstd) {
  int j = blockIdx.x, tid = threadIdx.x;
  float s = 0.f, s2 = 0.f;
  for (int i = tid; i < BN; i += 256) {
    float v = xc[(size_t)i * NCONTN + j];
    s += v;
    s2 += v * v;
  }
  __shared__ float r1[256], r2[256];
  r1[tid] = s;
  r2[tid] = s2;
  __syncthreads();
  for (int st = 128; st > 0; st >>= 1) {
    if (tid < st) { r1[tid] += r1[tid + st]; r2[tid] += r2[tid + st]; }
    __syncthreads();
  }
  if (tid == 0) {
    float m   = r1[0] * (1.0f / BN);
    float var = r2[0] * (1.0f / BN) - m * m;
    mu[j]   = m;
    rstd[j] = rsqrtf(var + 1e-5f);
  }
}

__global__ __launch_bounds__(256) void bnnorm_kernel(const float* __restrict__ xc,
                                                     const float* __restrict__ mu,
                                                     const float* __restrict__ rstd,
                                                     const float* __restrict__ g,
                                                     const float* __restrict__ b,
                                                     float* __restrict__ normed) {
  int idx = blockIdx.x * 256 + threadIdx.x;  // < B*NCONT = 262144
  int col = idx & 63;
  normed[idx] = (xc[idx] - mu[col]) * rstd[col] * g[col] + b[col];
}

// ---------------------------------------------------------------- big GEMMs
// C = A @ W (+ bias). One wave per 16x16 tile.
__global__ __launch_bounds__(128) void gemm_plain(const float* __restrict__ A, int lda,
                                                  const float* __restrict__ W,
                                                  const float* __restrict__ bias,
                                                  float* __restrict__ C, int N, int K,
                                                  int Ntiles) {
  int id   = blockIdx.x * 4 + (threadIdx.x >> 5);
  int lane = threadIdx.x & 31;
  int mt = id / Ntiles, nt = id % Ntiles;
  v8f acc = zero8();
  acc = wmma_acc_f32(A + (size_t)mt * 16 * lda, lda, W + nt * 16, N, K, acc, lane);
  int kh = lane >> 4, n = lane & 15;
  int col = nt * 16 + n;
  float bb = bias ? bias[col] : 0.0f;
  for (int r = 0; r < 8; ++r)
    C[(size_t)(mt * 16 + r + 8 * kh) * N + col] = acc[r] + bb;
}

// H = relu(concat(A1,A2) @ W + bias) * H   (in place; the gated-H of _gate_combine)
__global__ __launch_bounds__(128) void gemm_gate(const float* __restrict__ A1, int lda1,
                                                 int K1, const float* __restrict__ A2,
                                                 int lda2, int K2,
                                                 const float* __restrict__ W,
                                                 const float* __restrict__ bias,
                                                 float* __restrict__ H, int N,
                                                 int Ntiles) {
  int id   = blockIdx.x * 4 + (threadIdx.x >> 5);
  int lane = threadIdx.x & 31;
  int mt = id / Ntiles, nt = id % Ntiles;
  v8f acc = zero8();
  acc = wmma_acc_f32(A1 + (size_t)mt * 16 * lda1, lda1, W + nt * 16, N, K1, acc, lane);
  acc = wmma_acc_f32(A2 + (size_t)mt * 16 * lda2, lda2, W + (size_t)K1 * N + nt * 16, N,
                     K2, acc, lane);
  int kh = lane >> 4, n = lane & 15;
  int col = nt * 16 + n;
  for (int r = 0; r < 8; ++r) {
    size_t idx = (size_t)(mt * 16 + r + 8 * kh) * N + col;
    float g = acc[r] + bias[col];
    g = g > 0.f ? g : 0.f;
    H[idx] = g * H[idx];
  }
}

// H <- X + alpha*H,  alpha = clip(||x||/(||H||+1e-12) * 0.2, 0, 1)  (per row)
__global__ __launch_bounds__(256) void combine_kernel(const float* __restrict__ X,
                                                      float* __restrict__ H, int n) {
  int r = blockIdx.x, tid = threadIdx.x;
  const float* xr = X + (size_t)r * n;
  float* hr = H + (size_t)r * n;
  float sx = 0.f, sh = 0.f;
  for (int j = tid; j < n; j += 256) {
    float xv = xr[j], hv = hr[j];
    sx += xv * xv;
    sh += hv * hv;
  }
  __shared__ float rx[256], rh[256];
  rx[tid] = sx;
  rh[tid] = sh;
  __syncthreads();
  for (int st = 128; st > 0; st >>= 1) {
    if (tid < st) { rx[tid] += rx[tid + st]; rh[tid] += rh[tid + st]; }
    __syncthreads();
  }
  float alpha = fminf(fmaxf(sqrtf(rx[0]) / (sqrtf(rh[0]) + 1e-12f) * 0.2f, 0.f), 1.f);
  for (int j = tid; j < n; j += 256) hr[j] = xr[j] + alpha * hr[j];
}

// out = sigmoid(concat(c0, c1) @ fc_w + fc_b)
__global__ __launch_bounds__(256) void final_kernel(const float* __restrict__ c0,
                                                    const float* __restrict__ c1,
                                                    const float* __restrict__ fcw,
                                                    const float* __restrict__ fcb,
                                                    float* __restrict__ out) {
  int r = blockIdx.x, tid = threadIdx.x;
  float s = 0.f;
  const float* c0r = c0 + (size_t)r * CATFLT;
  for (int j = tid; j < CATFLT; j += 256) s += c0r[j] * fcw[j];
  const float* c1r = c1 + (size_t)r * NCONTN;
  if (tid < NCONTN) s += c1r[tid] * fcw[CATFLT + tid];
  __shared__ float red[256];
  red[tid] = s;
  __syncthreads();
  for (int st = 128; st > 0; st >>= 1) {
    if (tid < st) red[tid] += red[tid + st];
    __syncthreads();
  }
  if (tid == 0) {
    float logit = red[0] + fcb[0];
    out[r] = 1.0f / (1.0f + expf(-logit));
  }
}

// ---------------------------------------------------------------- launcher
extern "C" void kernel_launch(void* const* d_in, const int* in_sizes, int n_in,
                              void* d_out, int out_size, void* d_ws, size_t ws_size,
                              hipStream_t stream) {
  (void)in_sizes; (void)n_in; (void)out_size; (void)ws_size;
  const int*   x_categ = (const int*)d_in[0];
  const float* x_cont  = (const float*)d_in[1];
  const float* embed   = (const float*)d_in[2];
  const float* ln1_g   = (const float*)d_in[3];
  const float* ln1_b   = (const float*)d_in[4];
  const float* wqkv    = (const float*)d_in[5];
  const float* wo      = (const float*)d_in[6];
  const float* bo      = (const float*)d_in[7];
  const float* ln2_g   = (const float*)d_in[8];
  const float* ln2_b   = (const float*)d_in[9];
  const float* wff1    = (const float*)d_in[10];
  const float* bff1    = (const float*)d_in[11];
  const float* wff2    = (const float*)d_in[12];
  const float* bff2    = (const float*)d_in[13];
  const float* bn_g    = (const float*)d_in[14];
  const float* bn_b    = (const float*)d_in[15];
  const float* g0_w    = (const float*)d_in[16];
  const float* g0_b    = (const float*)d_in[17];
  const float* h0_w    = (const float*)d_in[18];
  const float* y1_w    = (const float*)d_in[19];
  const float* y1_b    = (const float*)d_in[20];
  const float* g1_w    = (const float*)d_in[21];
  const float* g1_b    = (const float*)d_in[22];
  const float* h1_w    = (const float*)d_in[23];
  const float* fc_w    = (const float*)d_in[24];
  const float* fc_b    = (const float*)d_in[25];
  float* out = (float*)d_out;

  float* ws     = (float*)d_ws;
  float* xbuf   = ws;                   // [B, 2048] == x tokens == flat_categ
  float* bufB   = ws + (size_t)BN * CATFLT;          // [B, 2048] hy0 -> H0 -> c0
  float* normed = bufB + (size_t)BN * CATFLT;        // [B, 64]
  float* y1buf  = normed + (size_t)BN * NCONTN;      // [B, 64]
  float* hy1buf = y1buf + (size_t)BN * NCONTN;       // [B, 64] hy1 -> H1 -> c1
  float* bnmu   = hy1buf + (size_t)BN * NCONTN;      // [64]
  float* bnrs   = bnmu + 64;                         // [64]

  // 1. embedding gather
  embed_kernel<<<(BN * NCN * DIMN) / 256, 256, 0, stream>>>(x_categ, embed, xbuf);

  // 2. 6-layer transformer, one sample per block, everything in LDS
  size_t smem = 20800 * sizeof(float);  // 83,200 B (<= 320 KB/WGP; ~3 WGs resident)
  transformer_kernel<<<BN, 128, smem, stream>>>(xbuf, ln1_g, ln1_b, wqkv, wo, bo, ln2_g,
                                                ln2_b, wff1, bff1, wff2, bff2);

  // 3. BatchNorm (train-mode batch stats)
  bnstats_kernel<<<NCONTN, 256, 0, stream>>>(x_cont, bnmu, bnrs);
  bnnorm_kernel<<<(BN * NCONTN) / 256, 256, 0, stream>>>(x_cont, bnmu, bnrs, bn_g, bn_b,
                                                         normed);

  // 4. hy0 = normed @ h0_w                      [4096,64]x[64,2048]
  gemm_plain<<<(256 * 128) / 4, 128, 0, stream>>>(normed, NCONTN, h0_w, nullptr, bufB,
                                                  CATFLT, NCONTN, 128);
  // 5. H0 = relu(concat(flat,normed)@g0_w + b) * hy0   (in place on bufB)
  gemm_gate<<<(256 * 128) / 4, 128, 0, stream>>>(xbuf, CATFLT, CATFLT, normed, NCONTN,
                                                 NCONTN, g0_w, g0_b, bufB, CATFLT, 128);
  // 6. c0 = flat + alpha0 * H0                   (in place on bufB)
  combine_kernel<<<BN, 256, 0, stream>>>(xbuf, bufB, CATFLT);

  // 7. y1 = flat @ y1_w + y1_b                   [4096,2048]x[2048,64]
  gemm_plain<<<(256 * 4) / 4, 128, 0, stream>>>(xbuf, CATFLT, y1_w, y1_b, y1buf, NCONTN,
                                                CATFLT, 4);
  // 8. hy1 = y1 @ h1_w
  gemm_plain<<<(256 * 4) / 4, 128, 0, stream>>>(y1buf, NCONTN, h1_w, nullptr, hy1buf,
                                                NCONTN, NCONTN, 4);
  // 9. H1 = relu(concat(normed,y1)@g1_w + b) * hy1     (in place on hy1buf)
  gemm_gate<<<(256 * 4) / 4, 128, 0, stream>>>(normed, NCONTN, NCONTN, y1buf, NCONTN,
                                               NCONTN, g1_w, g1_b, hy1buf, NCONTN, 4);
  // 10. c1 = normed + alpha1 * H1                (in place on hy1buf)
  combine_kernel<<<BN, 256, 0, stream>>>(normed, hy1buf, NCONTN);

  // 11. logits -> sigmoid
  final_kernel<<<BN, 256, 0, stream>>>(bufB, hy1buf, fc_w, fc_b, out);
}